// EGConvNet_80247168958859
// MI455X (gfx1250) — compile-verified
//
#include <hip/hip_runtime.h>

typedef float v2f __attribute__((ext_vector_type(2)));
typedef float v8f __attribute__((ext_vector_type(8)));

#define HID 256
#define NGRAPHS 2048
#define BN_EPS 1e-5f

// ---------- order-preserving float <-> uint for atomic max ----------
__device__ __forceinline__ unsigned encf(float x) {
    unsigned u = __float_as_uint(x);
    return (u & 0x80000000u) ? ~u : (u | 0x80000000u);
}
__device__ __forceinline__ float decf(unsigned u) {
    return (u & 0x80000000u) ? __uint_as_float(u ^ 0x80000000u) : __uint_as_float(~u);
}

// ---------- AtomEncoder: h[n,d] = sum_c emb[x[n,c]+off[c], d] ----------
__global__ void atom_encoder_kernel(const int* __restrict__ x,
                                    const float* __restrict__ emb,
                                    float* __restrict__ h, int N) {
    const int OFF[9] = {0, 119, 124, 136, 148, 158, 164, 170, 172};
    int n = blockIdx.x;
    int d = threadIdx.x;
    if (n >= N) return;
    float acc = 0.0f;
#pragma unroll
    for (int c = 0; c < 9; ++c) {
        int row = x[n * 9 + c] + OFF[c];
        acc += emb[(size_t)row * HID + d];
    }
    h[(size_t)n * HID + d] = acc;
}

// ---------- degree (self-loop augmented) ----------
__global__ void deg_init_kernel(float* __restrict__ deg, int N) {
    int i = blockIdx.x * blockDim.x + threadIdx.x;
    if (i < N) deg[i] = 1.0f;  // self loop
}
__global__ void deg_accum_kernel(const int* __restrict__ ei, float* __restrict__ deg, int E) {
    int e = blockIdx.x * blockDim.x + threadIdx.x;
    if (e < E) atomicAdd(&deg[ei[e]], 1.0f);  // ei row 0 = dst
}

// ---------- WMMA f32 GEMM: C[M,Ncol] = A[M,K] @ W[K,Ncol] (+bias) ----------
// One wave per 16x16 output tile, K stepped by 4 with V_WMMA_F32_16X16X4_F32.
__global__ void gemm_wmma_f32(const float* __restrict__ A, const float* __restrict__ W,
                              const float* __restrict__ bias, float* __restrict__ C,
                              int M, int K, int Ncol) {
    const int lane = threadIdx.x & 31;
    const int wavesPerBlock = blockDim.x >> 5;
    const int wid = blockIdx.x * wavesPerBlock + (threadIdx.x >> 5);
    const int tcols = Ncol >> 4;
    const int trows = M >> 4;
    if (wid >= tcols * trows) return;  // uniform per wave: EXEC stays all-ones for WMMA
    const int ty = wid / tcols;
    const int tx = wid - ty * tcols;
    const int row0 = ty << 4, col0 = tx << 4;
    const int half = lane >> 4;   // 0: K pair {0,1}; 1: K pair {2,3}
    const int l15 = lane & 15;

    const float* arow = A + (size_t)(row0 + l15) * K;  // this lane's A row
    const float* wcol = W + (col0 + l15);              // this lane's B column

    v8f c = {};
    for (int k = 0; k < K; k += 4) {
        const int ka = k + half * 2;
        v2f a, b;
        a.x = arow[ka];
        a.y = arow[ka + 1];
        b.x = wcol[(size_t)ka * Ncol];
        b.y = wcol[(size_t)(ka + 1) * Ncol];
        c = __builtin_amdgcn_wmma_f32_16x16x4_f32(false, a, false, b,
                                                  (short)0, c, false, false);
    }
    const float bv = bias ? bias[col0 + l15] : 0.0f;
    float* crow = C + (size_t)(row0 + half * 8) * Ncol + col0 + l15;
#pragma unroll
    for (int r = 0; r < 8; ++r)
        crow[(size_t)r * Ncol] = c[r] + bv;
}

// ---------- edge aggregation: one wave per edge, 4 floats/lane ----------
__global__ void agg_edges_kernel(const float* __restrict__ bases, const int* __restrict__ ei,
                                 int E, int N, float* __restrict__ asum,
                                 unsigned* __restrict__ amax) {
    long long gt = (long long)blockIdx.x * blockDim.x + threadIdx.x;
    int e = (int)(gt >> 5);
    if (e >= E + N) return;
    int lane = threadIdx.x & 31;
    int s, d;
    if (e < E) { d = ei[e]; s = ei[E + e]; }   // row0=dst, row1=src (adj_t)
    else       { d = s = e - E; }              // self loop
    const float4 v = *(const float4*)(bases + (size_t)s * 128 + lane * 4);
    float*    ps = asum + (size_t)d * 128 + lane * 4;
    unsigned* pm = amax + (size_t)d * 128 + lane * 4;
    atomicAdd(ps + 0, v.x); atomicAdd(ps + 1, v.y);
    atomicAdd(ps + 2, v.z); atomicAdd(ps + 3, v.w);
    atomicMax(pm + 0, encf(v.x)); atomicMax(pm + 1, encf(v.y));
    atomicMax(pm + 2, encf(v.z)); atomicMax(pm + 3, encf(v.w));
}

// ---------- per-node combine: out[n, h*32+d] = sum_k wts[n,h,k]*agg[n,k,d] ----------
__global__ void combine_kernel(const float* __restrict__ wts, const float* __restrict__ asum,
                               const unsigned* __restrict__ amax, const float* __restrict__ deg,
                               const float* __restrict__ cbias, float* __restrict__ outb, int N) {
    int n = blockIdx.x;
    int t = threadIdx.x;        // 0..255
    if (n >= N) return;
    int hh = t >> 5, d = t & 31;
    float invdeg = 1.0f / deg[n];
    const float* wr = wts + (size_t)n * 96 + hh * 12;  // k = a*4+b, a in {sum,mean,max}
    float acc = cbias[t];
#pragma unroll
    for (int b = 0; b < 4; ++b) {
        float sv = asum[(size_t)n * 128 + b * 32 + d];
        float mv = sv * invdeg;
        float xv = decf(amax[(size_t)n * 128 + b * 32 + d]);
        acc += wr[b] * sv + wr[4 + b] * mv + wr[8 + b] * xv;
    }
    outb[(size_t)n * 256 + t] = acc;
}

// ---------- BatchNorm: column stats (blockDim.x == C) ----------
__global__ void colstats_kernel(const float* __restrict__ X, int R, int C,
                                float* __restrict__ s1, float* __restrict__ s2,
                                int rowsPerBlock) {
    int c = threadIdx.x;
    int r0 = blockIdx.x * rowsPerBlock;
    int r1 = r0 + rowsPerBlock; if (r1 > R) r1 = R;
    float a = 0.0f, b = 0.0f;
    for (int r = r0; r < r1; ++r) {
        float v = X[(size_t)r * C + c];
        a += v; b += v * v;
    }
    atomicAdd(&s1[c], a);
    atomicAdd(&s2[c], b);
}
__global__ void bn_finalize_kernel(const float* __restrict__ s1, const float* __restrict__ s2,
                                   int R, int C, float* __restrict__ mu, float* __restrict__ rs) {
    int c = blockIdx.x * blockDim.x + threadIdx.x;
    if (c >= C) return;
    float m = s1[c] / (float)R;
    float v = s2[c] / (float)R - m * m;
    mu[c] = m;
    rs[c] = rsqrtf(v + BN_EPS);
}
__global__ void bn_relu_residual_kernel(const float* __restrict__ X, float* __restrict__ H,
                                        const float* __restrict__ mu, const float* __restrict__ rs,
                                        const float* __restrict__ gamma, const float* __restrict__ beta,
                                        long long total, int C) {
    long long i = (long long)blockIdx.x * blockDim.x + threadIdx.x;
    if (i >= total) return;
    int c = (int)(i % C);
    float v = (X[i] - mu[c]) * rs[c] * gamma[c] + beta[c];
    H[i] += fmaxf(v, 0.0f);
}
__global__ void bn_relu_inplace_kernel(float* __restrict__ X,
                                       const float* __restrict__ mu, const float* __restrict__ rs,
                                       const float* __restrict__ gamma, const float* __restrict__ beta,
                                       long long total, int C) {
    long long i = (long long)blockIdx.x * blockDim.x + threadIdx.x;
    if (i >= total) return;
    int c = (int)(i % C);
    float v = (X[i] - mu[c]) * rs[c] * gamma[c] + beta[c];
    X[i] = fmaxf(v, 0.0f);
}

// ---------- global mean pool ----------
__global__ void pool_accum_kernel(const float* __restrict__ H, const int* __restrict__ batch,
                                  float* __restrict__ gsum, float* __restrict__ gcnt, int N) {
    int n = blockIdx.x;
    int t = threadIdx.x;
    if (n >= N) return;
    int g = batch[n];
    atomicAdd(&gsum[(size_t)g * HID + t], H[(size_t)n * HID + t]);
    if (t == 0) atomicAdd(&gcnt[g], 1.0f);
}
__global__ void pool_div_kernel(float* __restrict__ gsum, const float* __restrict__ gcnt, int G) {
    long long i = (long long)blockIdx.x * blockDim.x + threadIdx.x;
    if (i >= (long long)G * HID) return;
    int g = (int)(i / HID);
    gsum[i] = gsum[i] / fmaxf(gcnt[g], 1.0f);
}

// ---------- final head: out[g] = z2[g,:] . w3 + b3 ----------
__global__ void head_kernel(const float* __restrict__ z, const float* __restrict__ w3,
                            const float* __restrict__ b3, float* __restrict__ out, int G) {
    int g = blockIdx.x * blockDim.x + threadIdx.x;
    if (g >= G) return;
    float acc = b3[0];
#pragma unroll 8
    for (int d = 0; d < 64; ++d) acc += z[(size_t)g * 64 + d] * w3[d];
    out[g] = acc;
}

extern "C" void kernel_launch(void* const* d_in, const int* in_sizes, int n_in,
                              void* d_out, int out_size, void* d_ws, size_t ws_size,
                              hipStream_t stream) {
    const float* atom_emb  = (const float*)d_in[0];
    const float* bases_W   = (const float*)d_in[1];
    const float* comb_W    = (const float*)d_in[2];
    const float* comb_b    = (const float*)d_in[3];
    const float* conv_bias = (const float*)d_in[4];
    const float* bn_gamma  = (const float*)d_in[5];
    const float* bn_beta   = (const float*)d_in[6];
    const float* w1 = (const float*)d_in[7];
    const float* g1 = (const float*)d_in[8];
    const float* b1 = (const float*)d_in[9];
    const float* w2 = (const float*)d_in[10];
    const float* g2 = (const float*)d_in[11];
    const float* b2 = (const float*)d_in[12];
    const float* w3 = (const float*)d_in[13];
    const float* b3 = (const float*)d_in[14];
    const int* x     = (const int*)d_in[15];
    const int* ei    = (const int*)d_in[16];
    const int* batch = (const int*)d_in[17];

    const int N = in_sizes[15] / 9;
    const int E = in_sizes[16] / 2;

    // ---- workspace partition (256B aligned) ----
    char* p = (char*)d_ws;
    auto alloc = [&](size_t bytes) -> void* {
        void* r = (void*)p;
        p += (bytes + 255) & ~(size_t)255;
        return r;
    };
    float*    h    = (float*)alloc((size_t)N * HID * 4);
    float*    outb = (float*)alloc((size_t)N * HID * 4);
    float*    bases= (float*)alloc((size_t)N * 128 * 4);
    float*    wts  = (float*)alloc((size_t)N * 96 * 4);
    float*    asum = (float*)alloc((size_t)N * 128 * 4);
    unsigned* amax = (unsigned*)alloc((size_t)N * 128 * 4);
    float*    deg  = (float*)alloc((size_t)N * 4);
    float*    s1   = (float*)alloc(HID * 4);
    float*    s2   = (float*)alloc(HID * 4);
    float*    mu   = (float*)alloc(HID * 4);
    float*    rs   = (float*)alloc(HID * 4);
    float*    gsum = (float*)alloc((size_t)NGRAPHS * HID * 4);
    float*    gcnt = (float*)alloc(NGRAPHS * 4);
    float*    z1   = (float*)alloc((size_t)NGRAPHS * 128 * 4);
    float*    z2   = (float*)alloc((size_t)NGRAPHS * 64 * 4);
    (void)ws_size; (void)n_in; (void)out_size;

    // ---- encoder + degrees ----
    atom_encoder_kernel<<<N, HID, 0, stream>>>(x, atom_emb, h, N);
    deg_init_kernel<<<(N + 255) / 256, 256, 0, stream>>>(deg, N);
    deg_accum_kernel<<<(E + 255) / 256, 256, 0, stream>>>(ei, deg, E);

    // ---- 4 EGConv layers ----
    for (int l = 0; l < 4; ++l) {
        const int tiles_b = (N / 16) * (128 / 16);
        gemm_wmma_f32<<<(tiles_b + 3) / 4, 128, 0, stream>>>(
            h, bases_W + (size_t)l * HID * 128, nullptr, bases, N, HID, 128);
        const int tiles_w = (N / 16) * (96 / 16);
        gemm_wmma_f32<<<(tiles_w + 3) / 4, 128, 0, stream>>>(
            h, comb_W + (size_t)l * HID * 96, comb_b + (size_t)l * 96, wts, N, HID, 96);

        hipMemsetAsync(asum, 0, (size_t)N * 128 * 4, stream);
        hipMemsetAsync(amax, 0, (size_t)N * 128 * 4, stream);  // 0 < encf(any finite)
        const long long tot = (long long)(E + N) * 32;
        agg_edges_kernel<<<(unsigned)((tot + 255) / 256), 256, 0, stream>>>(
            bases, ei, E, N, asum, amax);

        combine_kernel<<<N, HID, 0, stream>>>(wts, asum, amax, deg,
                                              conv_bias + (size_t)l * HID, outb, N);

        hipMemsetAsync(s1, 0, HID * 4, stream);
        hipMemsetAsync(s2, 0, HID * 4, stream);
        colstats_kernel<<<(N + 255) / 256, HID, 0, stream>>>(outb, N, HID, s1, s2, 256);
        bn_finalize_kernel<<<1, HID, 0, stream>>>(s1, s2, N, HID, mu, rs);
        const long long tot2 = (long long)N * HID;
        bn_relu_residual_kernel<<<(unsigned)((tot2 + 255) / 256), 256, 0, stream>>>(
            outb, h, mu, rs, bn_gamma + (size_t)l * HID, bn_beta + (size_t)l * HID, tot2, HID);
    }

    // ---- global mean pool ----
    hipMemsetAsync(gsum, 0, (size_t)NGRAPHS * HID * 4, stream);
    hipMemsetAsync(gcnt, 0, NGRAPHS * 4, stream);
    pool_accum_kernel<<<N, HID, 0, stream>>>(h, batch, gsum, gcnt, N);
    pool_div_kernel<<<(NGRAPHS * HID + 255) / 256, 256, 0, stream>>>(gsum, gcnt, NGRAPHS);

    // ---- MLP head ----
    gemm_wmma_f32<<<((NGRAPHS / 16) * (128 / 16) + 3) / 4, 128, 0, stream>>>(
        gsum, w1, nullptr, z1, NGRAPHS, HID, 128);
    hipMemsetAsync(s1, 0, 128 * 4, stream);
    hipMemsetAsync(s2, 0, 128 * 4, stream);
    colstats_kernel<<<(NGRAPHS + 255) / 256, 128, 0, stream>>>(z1, NGRAPHS, 128, s1, s2, 256);
    bn_finalize_kernel<<<1, 128, 0, stream>>>(s1, s2, NGRAPHS, 128, mu, rs);
    bn_relu_inplace_kernel<<<((NGRAPHS * 128) + 255) / 256, 256, 0, stream>>>(
        z1, mu, rs, g1, b1, (long long)NGRAPHS * 128, 128);

    gemm_wmma_f32<<<((NGRAPHS / 16) * (64 / 16) + 3) / 4, 128, 0, stream>>>(
        z1, w2, nullptr, z2, NGRAPHS, 128, 64);
    hipMemsetAsync(s1, 0, 64 * 4, stream);
    hipMemsetAsync(s2, 0, 64 * 4, stream);
    colstats_kernel<<<(NGRAPHS + 255) / 256, 64, 0, stream>>>(z2, NGRAPHS, 64, s1, s2, 256);
    bn_finalize_kernel<<<1, 64, 0, stream>>>(s1, s2, NGRAPHS, 64, mu, rs);
    bn_relu_inplace_kernel<<<((NGRAPHS * 64) + 255) / 256, 256, 0, stream>>>(
        z2, mu, rs, g2, b2, (long long)NGRAPHS * 64, 64);

    head_kernel<<<(NGRAPHS + 255) / 256, 256, 0, stream>>>(z2, w3, b3, (float*)d_out, NGRAPHS);
}